// CONN_36644660969834
// MI455X (gfx1250) — compile-verified
//
#include <hip/hip_runtime.h>
#include <hip/hip_bf16.h>

// ---------------------------------------------------------------------------
// 2-layer GCN forward for MI455X (gfx1250, wave32).
//   x0 = concat(node_emb, attri_emb)                [N=100000, D=128] f32
//   per layer: support = x @ W  (WMMA f32 16x16x4)
//              x' = scatter-add(edge_val * support[col] -> row) + b
//   12 gather + L2-normalize passes -> out [4, 3, 50000, 128]
// Bandwidth-bound (~3.6 GB total); GEMM done with V_WMMA_F32_16X16X4_F32.
// ---------------------------------------------------------------------------

#define N_NODE  50000
#define N_TOTAL 100000
#define DFEAT   128
#define E_EDGES 1600000
#define E_PAIR  50000

typedef float v2f __attribute__((ext_vector_type(2)));
typedef float v8f __attribute__((ext_vector_type(8)));

// ---------------------------------------------------------------------------
// Gather rows by index and L2-normalize (one wave32 per row, float4 per lane).
// Virtual concat handled via (xa, xb, split).
// ---------------------------------------------------------------------------
__global__ __launch_bounds__(256)
void gcn_gather_norm(const float* __restrict__ xa,
                     const float* __restrict__ xb,
                     int split,
                     const int* __restrict__ idx,
                     float* __restrict__ out,
                     int nrows) {
    int r    = (int)((blockIdx.x * blockDim.x + threadIdx.x) >> 5);
    int lane = (int)(threadIdx.x & 31);
    if (r >= nrows) return;

    int j = idx[r];
    const float* src = (j < split) ? (xa + (size_t)j * DFEAT)
                                   : (xb + (size_t)(j - split) * DFEAT);
    float4 v = ((const float4*)src)[lane];

    float ss = v.x * v.x + v.y * v.y + v.z * v.z + v.w * v.w;
    // wave32 butterfly reduction
    #pragma unroll
    for (int m = 16; m >= 1; m >>= 1)
        ss += __shfl_xor(ss, m, 32);

    float inv = 1.0f / fmaxf(sqrtf(ss), 1e-12f);
    float4 o;
    o.x = v.x * inv; o.y = v.y * inv; o.z = v.z * inv; o.w = v.w * inv;
    ((float4*)(out + (size_t)r * DFEAT))[lane] = o;
}

// ---------------------------------------------------------------------------
// support = x @ W via V_WMMA_F32_16X16X4_F32.
// One wave per 16x16 output tile; K accumulated in steps of 4 (32 WMMAs).
// A-frag (16x4 f32): lane L holds A[L&15][kbase + (L>>4)*2 + {0,1}]
// B-frag (4x16 f32): lane L holds W[kbase + (L>>4)*2 + {0,1}][tileN*16 + (L&15)]
// C/D    (16x16 f32): VGPR v, lane L -> M = (L>>4)*8 + v, Ncol = L&15
// M = 100000 and the grid match exactly, so EXEC is all-1s for every WMMA.
// ---------------------------------------------------------------------------
__global__ __launch_bounds__(256)
void gcn_gemm_wmma(const float* __restrict__ xa,
                   const float* __restrict__ xb,
                   int split,
                   const float* __restrict__ W,
                   float* __restrict__ outp,
                   int M) {
    const int tilesN = DFEAT / 16;                       // 8
    int wave = (int)((blockIdx.x * blockDim.x + threadIdx.x) >> 5);
    int lane = (int)(threadIdx.x & 31);
    int tileM = wave / tilesN;
    int tileN = wave % tilesN;
    if (tileM * 16 >= M) return;                         // wave-uniform

    int arow = tileM * 16 + (lane & 15);
    const float* Arow = (arow < split) ? (xa + (size_t)arow * DFEAT)
                                       : (xb + (size_t)(arow - split) * DFEAT);
    int khalf = (lane >> 4) * 2;                         // 0 or 2
    int bcol  = tileN * 16 + (lane & 15);

    v8f acc = {0.f, 0.f, 0.f, 0.f, 0.f, 0.f, 0.f, 0.f};

    #pragma unroll 4
    for (int k = 0; k < DFEAT; k += 4) {
        v2f a = *(const v2f*)(Arow + k + khalf);
        v2f b;
        b.x = W[(size_t)(k + khalf + 0) * DFEAT + bcol];
        b.y = W[(size_t)(k + khalf + 1) * DFEAT + bcol];
        acc = __builtin_amdgcn_wmma_f32_16x16x4_f32(
                  /*neg_a=*/false, a, /*neg_b=*/false, b,
                  /*c_mod=*/(short)0, acc,
                  /*reuse_a=*/false, /*reuse_b=*/false);
    }

    int crow0 = tileM * 16 + (lane >> 4) * 8;
    int ccol  = tileN * 16 + (lane & 15);
    #pragma unroll
    for (int v = 0; v < 8; ++v)
        outp[(size_t)(crow0 + v) * DFEAT + ccol] = acc[v];
}

// ---------------------------------------------------------------------------
// Seed the segment-sum accumulator with the broadcast bias (x = 0 + b).
// ---------------------------------------------------------------------------
__global__ __launch_bounds__(256)
void gcn_init_bias(float* __restrict__ x, const float* __restrict__ b, int n) {
    int i = (int)(blockIdx.x * blockDim.x + threadIdx.x);
    if (i < n) x[i] = b[i & (DFEAT - 1)];
}

// ---------------------------------------------------------------------------
// Edge-parallel SpMM scatter: one wave32 per edge.
// Lane loads float4 of support[col], scales by edge_val, atomic-adds into
// x[row]. Atomics land in L2 (51.2 MB accumulator << 192 MB L2).
// ---------------------------------------------------------------------------
__global__ __launch_bounds__(256)
void gcn_spmm_scatter(const float* __restrict__ support,
                      const float* __restrict__ val,
                      const int* __restrict__ rows,
                      const int* __restrict__ cols,
                      float* __restrict__ x,
                      int nedges) {
    int e    = (int)((blockIdx.x * blockDim.x + threadIdx.x) >> 5);
    int lane = (int)(threadIdx.x & 31);
    if (e >= nedges) return;

    int   r = rows[e];
    int   c = cols[e];
    float v = val[e];

    float4 m = ((const float4*)(support + (size_t)c * DFEAT))[lane];
    float* o = x + (size_t)r * DFEAT + (size_t)lane * 4;
    atomicAdd(o + 0, v * m.x);
    atomicAdd(o + 1, v * m.y);
    atomicAdd(o + 2, v * m.z);
    atomicAdd(o + 3, v * m.w);
}

// ---------------------------------------------------------------------------
// Launch: hop0 gathers -> [gemm, init, scatter, gathers] x 2 layers.
// Workspace: [support | xbuf], 2 x 51.2 MB. xbuf is reused as the layer-2
// accumulator only after the layer-2 GEMM (its last reader) in stream order.
// ---------------------------------------------------------------------------
extern "C" void kernel_launch(void* const* d_in, const int* in_sizes, int n_in,
                              void* d_out, int out_size, void* d_ws, size_t ws_size,
                              hipStream_t stream) {
    const float* node_emb  = (const float*)d_in[0];
    const float* attri_emb = (const float*)d_in[1];
    const float* W1        = (const float*)d_in[2];
    const float* b1        = (const float*)d_in[3];
    const float* W2        = (const float*)d_in[4];
    const float* b2        = (const float*)d_in[5];
    const float* edge_val  = (const float*)d_in[6];
    const int*   edge_row  = (const int*)d_in[7];
    const int*   edge_col  = (const int*)d_in[8];
    const int*   pidx[4]   = {(const int*)d_in[9],  (const int*)d_in[10],
                              (const int*)d_in[11], (const int*)d_in[12]};
    float* out = (float*)d_out;

    float* support = (float*)d_ws;                       // 51.2 MB
    float* xbuf    = support + (size_t)N_TOTAL * DFEAT;  // 51.2 MB

    const int nElems     = N_TOTAL * DFEAT;              // 12.8M
    const int gridGather = (E_PAIR * 32 + 255) / 256;    // 6250
    const int gridGemm   = ((N_TOTAL / 16) * (DFEAT / 16) * 32) / 256; // 6250
    const int gridInit   = (nElems + 255) / 256;         // 50000
    const int gridSpmm   = (E_EDGES * 32 + 255) / 256;   // 200000

    const size_t hopStride = (size_t)E_PAIR * DFEAT;     // per (stream,hop) slab

    // ---- hop 0: normalize rows of virtual concat(node_emb, attri_emb) ----
    for (int s = 0; s < 4; ++s)
        gcn_gather_norm<<<gridGather, 256, 0, stream>>>(
            node_emb, attri_emb, N_NODE, pidx[s],
            out + ((size_t)s * 3 + 0) * hopStride, E_PAIR);

    // ---- layer 1 ----
    gcn_gemm_wmma<<<gridGemm, 256, 0, stream>>>(
        node_emb, attri_emb, N_NODE, W1, support, N_TOTAL);
    gcn_init_bias<<<gridInit, 256, 0, stream>>>(xbuf, b1, nElems);
    gcn_spmm_scatter<<<gridSpmm, 256, 0, stream>>>(
        support, edge_val, edge_row, edge_col, xbuf, E_EDGES);
    for (int s = 0; s < 4; ++s)
        gcn_gather_norm<<<gridGather, 256, 0, stream>>>(
            xbuf, xbuf, N_TOTAL, pidx[s],
            out + ((size_t)s * 3 + 1) * hopStride, E_PAIR);

    // ---- layer 2 (xbuf safely recycled after its GEMM read) ----
    gcn_gemm_wmma<<<gridGemm, 256, 0, stream>>>(
        xbuf, xbuf, N_TOTAL, W2, support, N_TOTAL);
    gcn_init_bias<<<gridInit, 256, 0, stream>>>(xbuf, b2, nElems);
    gcn_spmm_scatter<<<gridSpmm, 256, 0, stream>>>(
        support, edge_val, edge_row, edge_col, xbuf, E_EDGES);
    for (int s = 0; s < 4; ++s)
        gcn_gather_norm<<<gridGather, 256, 0, stream>>>(
            xbuf, xbuf, N_TOTAL, pidx[s],
            out + ((size_t)s * 3 + 2) * hopStride, E_PAIR);
}